// VectorCollapseEngine_31207232373226
// MI455X (gfx1250) — compile-verified
//
#include <hip/hip_runtime.h>
#include <math.h>

typedef float v2f __attribute__((ext_vector_type(2)));
typedef float v8f __attribute__((ext_vector_type(8)));

#define DIM      4096
#define BATCH    8192
#define NLAYERS  6
#define KSPLIT   8                 // K-dimension split for dots pass
#define SEG      (DIM / KSPLIT)    // 512 columns per wave

// ------------------------------------------------------------------
// Kernel 0: anchor norms + Gram matrix of normalized anchors.
// hdr[0..2] = 1/max(||anchor_k||, 1e-12);  hdr[3..11] = G row-major.
// ------------------------------------------------------------------
__global__ __launch_bounds__(256) void vce_setup(const float* __restrict__ ae,
                                                 const float* __restrict__ ac,
                                                 const float* __restrict__ an,
                                                 float* __restrict__ hdr) {
    __shared__ float red[6][256];
    float acc[6] = {0.f, 0.f, 0.f, 0.f, 0.f, 0.f};
    for (int i = threadIdx.x; i < DIM; i += 256) {
        float e = ae[i], c = ac[i], n = an[i];
        acc[0] += e * e; acc[1] += c * c; acc[2] += n * n;
        acc[3] += e * c; acc[4] += e * n; acc[5] += c * n;
    }
    for (int q = 0; q < 6; ++q) red[q][threadIdx.x] = acc[q];
    __syncthreads();
    for (int off = 128; off > 0; off >>= 1) {
        if (threadIdx.x < off)
            for (int q = 0; q < 6; ++q)
                red[q][threadIdx.x] += red[q][threadIdx.x + off];
        __syncthreads();
    }
    if (threadIdx.x == 0) {
        double ss[3] = {red[0][0], red[1][0], red[2][0]};
        double p01 = red[3][0], p02 = red[4][0], p12 = red[5][0];
        double inv[3];
        for (int k = 0; k < 3; ++k) inv[k] = 1.0 / fmax(sqrt(ss[k]), 1e-12);
        hdr[0] = (float)inv[0]; hdr[1] = (float)inv[1]; hdr[2] = (float)inv[2];
        double G[3][3];
        G[0][0] = ss[0]*inv[0]*inv[0]; G[1][1] = ss[1]*inv[1]*inv[1]; G[2][2] = ss[2]*inv[2]*inv[2];
        G[0][1] = G[1][0] = p01*inv[0]*inv[1];
        G[0][2] = G[2][0] = p02*inv[0]*inv[2];
        G[1][2] = G[2][1] = p12*inv[1]*inv[2];
        for (int j = 0; j < 3; ++j)
            for (int k = 0; k < 3; ++k)
                hdr[3 + j*3 + k] = (float)G[j][k];
    }
}

// ------------------------------------------------------------------
// Kernel 1: per-row dots via V_WMMA_F32_16X16X4_F32.
// Wave = 16 rows x SEG columns. D[M,N] = sum_K h[M,K]*dir_N[K], N=0..2.
// s^2 accumulated in VALU (lane L + lane L+16 cover all K of row L&15).
// Partials: part[s][b*4 + {a0,a1,a2,s2}], reduced deterministically later.
// ------------------------------------------------------------------
__global__ __launch_bounds__(256) void vce_dots(const float* __restrict__ h0,
                                                const float* __restrict__ ae,
                                                const float* __restrict__ ac,
                                                const float* __restrict__ an,
                                                const float* __restrict__ hdr,
                                                float* __restrict__ part) {
    const int lane = threadIdx.x & 31;
    const int wave = (blockIdx.x << 3) + (threadIdx.x >> 5); // 8 waves/block
    const int seg  = wave & (KSPLIT - 1);
    const int rowBase = (wave / KSPLIT) << 4;
    const int n  = lane & 15;      // N (anchor) for B; row for A
    const int hi = lane >> 4;      // half select: K = 2*hi + {0,1}
    const int colBase = seg * SEG;

    const float* aptr = h0 + (size_t)(rowBase + n) * DIM + colBase + hi * 2;
    const float* bsrc = (n == 0) ? ae : (n == 1) ? ac : (n == 2) ? an : ae;
    const float  bscale = (n < 3) ? hdr[n] : 0.0f;   // fold 1/||anchor|| into B
    const float* bptr = bsrc + colBase + hi * 2;

    v8f c0 = {0.f,0.f,0.f,0.f,0.f,0.f,0.f,0.f};
    v8f c1 = {0.f,0.f,0.f,0.f,0.f,0.f,0.f,0.f};
    float s2a = 0.f, s2b = 0.f;

    for (int kb = 0; kb < SEG; kb += 8) {
        v2f a0 = *(const v2f*)(aptr + kb);
        v2f a1 = *(const v2f*)(aptr + kb + 4);
        v2f b0 = *(const v2f*)(bptr + kb);
        v2f b1 = *(const v2f*)(bptr + kb + 4);
        b0 *= bscale;
        b1 *= bscale;
        s2a += a0.x*a0.x + a0.y*a0.y;
        s2b += a1.x*a1.x + a1.y*a1.y;
        c0 = __builtin_amdgcn_wmma_f32_16x16x4_f32(false, a0, false, b0,
                                                   (short)0, c0, false, false);
        c1 = __builtin_amdgcn_wmma_f32_16x16x4_f32(false, a1, false, b1,
                                                   (short)0, c1, false, false);
    }

    float s2 = s2a + s2b;
    s2 += __shfl_xor(s2, 16, 32);  // combine K-halves of this row

    float* p = part + (size_t)seg * BATCH * 4;
    if (n < 3) {
        // C/D layout: VGPR j -> M = j + 8*hi, N = n
        #pragma unroll
        for (int j = 0; j < 8; ++j)
            p[(size_t)(rowBase + hi * 8 + j) * 4 + n] = c0[j] + c1[j];
    }
    if (hi == 0)
        p[(size_t)(rowBase + n) * 4 + 3] = s2;
}

// ------------------------------------------------------------------
// Kernel 2: per-row 6-layer scalar recurrence (double precision).
// Reduces K-split partials deterministically, emits traces + coeffs.
// h(t) = A*h0 + sum_k Bc[k]*d_k ; only (s^2, a_k) evolve.
// ------------------------------------------------------------------
__global__ __launch_bounds__(256) void vce_recur(const float* __restrict__ hdr,
                                                 const float* __restrict__ part,
                                                 float* __restrict__ coeffs,
                                                 float* __restrict__ out) {
    int b = blockIdx.x * blockDim.x + threadIdx.x;
    if (b >= BATCH) return;

    double invn[3], G[3][3];
    for (int k = 0; k < 3; ++k) invn[k] = (double)hdr[k];
    for (int j = 0; j < 3; ++j)
        for (int k = 0; k < 3; ++k) G[j][k] = (double)hdr[3 + j*3 + k];

    double a[3] = {0.0, 0.0, 0.0}, s2 = 0.0;
    for (int s = 0; s < KSPLIT; ++s) {   // fixed order -> deterministic
        const float* p = part + (size_t)s * BATCH * 4 + (size_t)b * 4;
        a[0] += (double)p[0]; a[1] += (double)p[1]; a[2] += (double)p[2];
        s2   += (double)p[3];
    }

    const double strengths[3] = {0.1, 0.1, 0.05};
    double A = 1.0, Bc[3] = {0.0, 0.0, 0.0};

    float* aligns = out + (size_t)BATCH * DIM;
    float* divs   = aligns + (size_t)NLAYERS * BATCH * 3;
    float* tens   = divs   + (size_t)NLAYERS * BATCH * 3;

    for (int l = 0; l < NLAYERS; ++l) {
        double s = sqrt(fmax(s2, 0.0));
        double sden = fmax(s, 1e-12);
        double alignv[3], divv[3], beta[3], bsum = 0.0;
        for (int k = 0; k < 3; ++k) {
            alignv[k] = a[k] / sden;
            divv[k]   = 1.0 - alignv[k];
            double r  = sqrt(fmax(s2 - 2.0 * a[k] + G[k][k], 0.0));
            beta[k]   = strengths[k] * divv[k] / fmax(r, 1e-12);
            bsum     += beta[k];
        }
        size_t tb = (size_t)l * BATCH * 3 + (size_t)b * 3;
        for (int k = 0; k < 3; ++k) {
            aligns[tb + k] = (float)alignv[k];
            divs[tb + k]   = (float)divv[k];
            tens[tb + k]   = (float)fmax(divv[k], 0.0);
        }
        double alpha = 1.0 - bsum;
        double cross = 0.0, quad = 0.0;
        for (int k = 0; k < 3; ++k) {
            cross += beta[k] * a[k];
            for (int j = 0; j < 3; ++j) quad += beta[j] * beta[k] * G[j][k];
        }
        double s2n = alpha * alpha * s2 + 2.0 * alpha * cross + quad;
        double an_[3];
        for (int k = 0; k < 3; ++k) {
            double t = alpha * a[k];
            for (int j = 0; j < 3; ++j) t += beta[j] * G[j][k];
            an_[k] = t;
        }
        A *= alpha;
        for (int k = 0; k < 3; ++k) { Bc[k] = alpha * Bc[k] + beta[k]; a[k] = an_[k]; }
        s2 = s2n;
        double hn = sqrt(fmax(s2, 0.0));
        if (hn > 10.0) {
            double m = 10.0 / (hn + 1e-8);
            A *= m; s2 *= m * m;
            for (int k = 0; k < 3; ++k) { Bc[k] *= m; a[k] *= m; }
        }
    }
    // fold anchor inverse-norms so combine can use raw anchors
    coeffs[(size_t)b*4 + 0] = (float)A;
    coeffs[(size_t)b*4 + 1] = (float)(Bc[0] * invn[0]);
    coeffs[(size_t)b*4 + 2] = (float)(Bc[1] * invn[1]);
    coeffs[(size_t)b*4 + 3] = (float)(Bc[2] * invn[2]);
}

// ------------------------------------------------------------------
// Kernel 3: h_final = A*h0 + B0*ae + B1*ac + B2*an   (float4 streams;
// h0 should still be L2-resident from the dots pass: 128MB < 192MB L2)
// ------------------------------------------------------------------
__global__ __launch_bounds__(256) void vce_combine(const float* __restrict__ h0,
                                                   const float* __restrict__ ae,
                                                   const float* __restrict__ ac,
                                                   const float* __restrict__ an,
                                                   const float* __restrict__ coeffs,
                                                   float* __restrict__ out) {
    size_t idx = (size_t)blockIdx.x * blockDim.x + threadIdx.x;  // float4 index
    if (idx >= (size_t)BATCH * (DIM / 4)) return;
    int b  = (int)(idx / (DIM / 4));
    int d4 = (int)(idx % (DIM / 4));
    float4 cf = *(const float4*)(coeffs + (size_t)b * 4);
    float4 h  = *(const float4*)(h0 + idx * 4);
    float4 e  = *(const float4*)(ae + (size_t)d4 * 4);
    float4 c  = *(const float4*)(ac + (size_t)d4 * 4);
    float4 n  = *(const float4*)(an + (size_t)d4 * 4);
    float4 r;
    r.x = cf.x*h.x + cf.y*e.x + cf.z*c.x + cf.w*n.x;
    r.y = cf.x*h.y + cf.y*e.y + cf.z*c.y + cf.w*n.y;
    r.z = cf.x*h.z + cf.y*e.z + cf.z*c.z + cf.w*n.z;
    r.w = cf.x*h.w + cf.y*e.w + cf.z*c.w + cf.w*n.w;
    *(float4*)(out + idx * 4) = r;
}

extern "C" void kernel_launch(void* const* d_in, const int* in_sizes, int n_in,
                              void* d_out, int out_size, void* d_ws, size_t ws_size,
                              hipStream_t stream) {
    (void)in_sizes; (void)n_in; (void)out_size; (void)ws_size;
    const float* h0 = (const float*)d_in[0];
    const float* ae = (const float*)d_in[1];
    const float* ac = (const float*)d_in[2];
    const float* an = (const float*)d_in[3];
    float* out = (float*)d_out;
    float* ws  = (float*)d_ws;

    float* hdr    = ws;                                   // 16 floats
    float* part   = ws + 16;                              // KSPLIT*BATCH*4 floats
    float* coeffs = part + (size_t)KSPLIT * BATCH * 4;    // BATCH*4 floats

    vce_setup<<<1, 256, 0, stream>>>(ae, ac, an, hdr);

    int dotWaves  = (BATCH / 16) * KSPLIT;                // 4096 waves
    vce_dots<<<dotWaves / 8, 256, 0, stream>>>(h0, ae, ac, an, hdr, part);

    vce_recur<<<(BATCH + 255) / 256, 256, 0, stream>>>(hdr, part, coeffs, out);

    size_t vec4 = (size_t)BATCH * (DIM / 4);
    vce_combine<<<(unsigned)((vec4 + 255) / 256), 256, 0, stream>>>(h0, ae, ac, an, coeffs, out);
}